// MixtureOfExperts_81484119540337
// MI455X (gfx1250) — compile-verified
//
#include <hip/hip_runtime.h>
#include <hip/hip_bf16.h>

// ---------------- problem constants ----------------
#define N_TOKENS   8192
#define IN_DIM     1024
#define OUT_DIM    1024
#define NEXP       8
#define HID        64
#define TILE_M     32
#define ENTRY_CAP  (2 * N_TOKENS + NEXP * TILE_M)   // 16640, multiple of 32
#define MAX_TILES  (ENTRY_CAP / TILE_M)             // 520

typedef __attribute__((ext_vector_type(2))) float v2f;
typedef __attribute__((ext_vector_type(8))) float v8f;

// ---------------------------------------------------------------------------
// 1) Gating: h = relu(x@Wg1.T + bg1); s = h@Wg2.T + bg2; softmax; top-2.
//    One block (64 threads) per token.
// ---------------------------------------------------------------------------
__global__ void moe_gate_kernel(const float* __restrict__ x,
                                const float* __restrict__ Wg1,
                                const float* __restrict__ bg1,
                                const float* __restrict__ Wg2,
                                const float* __restrict__ bg2,
                                float* __restrict__ gate_out,     // [N, 8]
                                int*   __restrict__ top2_out,     // [N, 2]
                                int*   __restrict__ ws_top2i,     // [N, 2]
                                float* __restrict__ ws_top2w,     // [N, 2]
                                int*   __restrict__ counts)       // [8]
{
    __shared__ float sx[IN_DIM];
    __shared__ float sh[HID];
    __shared__ float sg[NEXP];

    const int n = blockIdx.x;
    const int t = threadIdx.x;           // 0..63
    const float* xr = x + (size_t)n * IN_DIM;

    for (int i = t; i < IN_DIM / 4; i += 64)
        ((float4*)sx)[i] = ((const float4*)xr)[i];
    __syncthreads();

    const float* w = Wg1 + (size_t)t * IN_DIM;
    float acc = 0.f;
    #pragma unroll 4
    for (int k = 0; k < IN_DIM; k += 4) {
        float4 a = *(const float4*)(sx + k);
        float4 b = *(const float4*)(w + k);
        acc += a.x * b.x + a.y * b.y + a.z * b.z + a.w * b.w;
    }
    sh[t] = fmaxf(acc + bg1[t], 0.f);
    __syncthreads();

    if (t < NEXP) {
        const float* w2 = Wg2 + (size_t)t * HID;
        float s = bg2[t];
        #pragma unroll
        for (int k = 0; k < HID; ++k) s += sh[k] * w2[k];
        sg[t] = s;
    }
    __syncthreads();

    if (t == 0) {
        float m = sg[0];
        #pragma unroll
        for (int e = 1; e < NEXP; ++e) m = fmaxf(m, sg[e]);
        float p[NEXP], den = 0.f;
        #pragma unroll
        for (int e = 0; e < NEXP; ++e) { p[e] = expf(sg[e] - m); den += p[e]; }
        const float inv = 1.f / den;
        int i1 = 0; float v1 = -1.f;
        #pragma unroll
        for (int e = 0; e < NEXP; ++e) {
            p[e] *= inv;
            gate_out[(size_t)n * NEXP + e] = p[e];
            if (p[e] > v1) { v1 = p[e]; i1 = e; }       // first max wins (jax tie-break)
        }
        int i2 = 0; float v2 = -1.f;
        #pragma unroll
        for (int e = 0; e < NEXP; ++e) {
            if (e == i1) continue;
            if (p[e] > v2) { v2 = p[e]; i2 = e; }
        }
        top2_out[n * 2 + 0] = i1;  top2_out[n * 2 + 1] = i2;
        ws_top2i[n * 2 + 0] = i1;  ws_top2i[n * 2 + 1] = i2;
        ws_top2w[n * 2 + 0] = v1;  ws_top2w[n * 2 + 1] = v2;
        atomicAdd(&counts[i1], 1);
        atomicAdd(&counts[i2], 1);
    }
}

// ---------------------------------------------------------------------------
// 2) Padded prefix-sum of per-expert counts (segments are multiples of 32).
// ---------------------------------------------------------------------------
__global__ void moe_offsets_kernel(const int* __restrict__ counts,
                                   int* __restrict__ off)   // [9]
{
    if (threadIdx.x == 0 && blockIdx.x == 0) {
        int acc = 0;
        for (int e = 0; e < NEXP; ++e) {
            off[e] = acc;
            acc += (counts[e] + (TILE_M - 1)) & ~(TILE_M - 1);
        }
        off[NEXP] = acc;
    }
}

// ---------------------------------------------------------------------------
// 3) Scatter tokens into per-expert segments (entry arrays are pre-zeroed,
//    so pad slots hold token 0 / weight 0 -> contribute exact +0.0).
// ---------------------------------------------------------------------------
__global__ void moe_scatter_kernel(const int*   __restrict__ ws_top2i,
                                   const float* __restrict__ ws_top2w,
                                   const int*   __restrict__ off,
                                   int*   __restrict__ cursors,
                                   int*   __restrict__ entryTok,
                                   float* __restrict__ entryW)
{
    const int n = blockIdx.x * blockDim.x + threadIdx.x;
    if (n >= N_TOKENS) return;
    #pragma unroll
    for (int s = 0; s < 2; ++s) {
        const int e   = ws_top2i[n * 2 + s];
        const int pos = off[e] + atomicAdd(&cursors[e], 1);
        entryTok[pos] = n;
        entryW[pos]   = ws_top2w[n * 2 + s];
    }
}

// ---------------------------------------------------------------------------
// 4) Routed expert GEMM with V_WMMA_F32_16X16X4_F32.
//    Block = 128 threads (4 waves). One 32-entry tile x 256 output cols.
//    Each wave: 32 tokens x 64 cols = 8 accumulators (two M halves x 4 col
//    tiles). Per K-step of 4: 2 A loads + 4 B loads feed 8 WMMAs
//    (10.7 FLOP/byte from cache, vs 6.4 for the 16-token tile).
//    A layout (16x4 f32): lanes 0-15 hold K=0,1; lanes 16-31 hold K=2,3
//      -> one contiguous float2 load per lane per K-step per M-half.
//    B layout (4x16 f32): VGPR0 = row K=khalf, VGPR1 = K=khalf+1, N = lane&15.
// ---------------------------------------------------------------------------
__global__ void moe_expert_wmma_kernel(const float* __restrict__ x,
                                       const float* __restrict__ We,
                                       const float* __restrict__ be,
                                       const int*   __restrict__ off,      // [9]
                                       const int*   __restrict__ entryTok,
                                       const float* __restrict__ entryW,
                                       float*       __restrict__ out)      // [N, OUT_DIM]
{
    __shared__ int   s_tok[TILE_M];
    __shared__ float s_w[TILE_M];
    __shared__ int   s_e;

    const int tile = blockIdx.x;
    const int base = tile * TILE_M;
    const int t    = threadIdx.x;            // 0..127

    if (t == 0) {
        int e = 0;
        #pragma unroll
        for (int i = 0; i < NEXP; ++i)
            if (base >= off[i]) e = i;       // last segment starting at/before base
        s_e = e;
    }
    if (t < TILE_M) {
        s_tok[t] = entryTok[base + t];
        s_w[t]   = entryW[base + t];
    }
    __syncthreads();

    const int e       = s_e;
    const int lane    = t & 31;
    const int wave    = t >> 5;
    const int colBase = blockIdx.y * 256 + wave * 64;
    const int mlo     = lane & 15;
    const int khalf   = (lane >> 4) << 1;    // 0 or 2

    // A fragments: M-half 0 = tile rows 0..15, M-half 1 = rows 16..31
    const float* xr0 = x + (size_t)s_tok[mlo]      * IN_DIM + khalf;
    const float* xr1 = x + (size_t)s_tok[mlo + 16] * IN_DIM + khalf;
    const float* wb  = We + (size_t)e * OUT_DIM * IN_DIM;
    const float* w0  = wb + (size_t)(colBase + mlo) * IN_DIM + khalf;
    const float* w1  = w0 + 16 * IN_DIM;
    const float* w2  = w0 + 32 * IN_DIM;
    const float* w3  = w0 + 48 * IN_DIM;

    v8f c00 = {}, c01 = {}, c02 = {}, c03 = {};
    v8f c10 = {}, c11 = {}, c12 = {}, c13 = {};

    #pragma unroll 2
    for (int k = 0; k < IN_DIM; k += 4) {
        const v2f a0 = *(const v2f*)(xr0 + k);
        const v2f a1 = *(const v2f*)(xr1 + k);
        const v2f b0 = *(const v2f*)(w0 + k);
        const v2f b1 = *(const v2f*)(w1 + k);
        const v2f b2 = *(const v2f*)(w2 + k);
        const v2f b3 = *(const v2f*)(w3 + k);
        c00 = __builtin_amdgcn_wmma_f32_16x16x4_f32(false, a0, false, b0, (short)0, c00, false, false);
        c01 = __builtin_amdgcn_wmma_f32_16x16x4_f32(false, a0, false, b1, (short)0, c01, false, false);
        c02 = __builtin_amdgcn_wmma_f32_16x16x4_f32(false, a0, false, b2, (short)0, c02, false, false);
        c03 = __builtin_amdgcn_wmma_f32_16x16x4_f32(false, a0, false, b3, (short)0, c03, false, false);
        c10 = __builtin_amdgcn_wmma_f32_16x16x4_f32(false, a1, false, b0, (short)0, c10, false, false);
        c11 = __builtin_amdgcn_wmma_f32_16x16x4_f32(false, a1, false, b1, (short)0, c11, false, false);
        c12 = __builtin_amdgcn_wmma_f32_16x16x4_f32(false, a1, false, b2, (short)0, c12, false, false);
        c13 = __builtin_amdgcn_wmma_f32_16x16x4_f32(false, a1, false, b3, (short)0, c13, false, false);
    }

    // D layout: lane L, VGPR r -> M = r + (L>=16 ? 8 : 0), N = L & 15
    const int mhi = (lane >> 4) << 3;        // 0 or 8
    const v8f accLo[4] = { c00, c01, c02, c03 };
    const v8f accHi[4] = { c10, c11, c12, c13 };
    #pragma unroll
    for (int j = 0; j < 4; ++j) {
        const int   nn   = colBase + j * 16 + mlo;
        const float bias = be[(size_t)e * OUT_DIM + nn];
        #pragma unroll
        for (int r = 0; r < 8; ++r) {
            const int m0 = r + mhi;          // M-half 0: tile rows 0..15
            const int m1 = m0 + 16;          // M-half 1: tile rows 16..31
            const float v0 = s_w[m0] * (accLo[j][r] + bias);
            const float v1 = s_w[m1] * (accHi[j][r] + bias);
            unsafeAtomicAdd(&out[(size_t)s_tok[m0] * OUT_DIM + nn], v0);
            unsafeAtomicAdd(&out[(size_t)s_tok[m1] * OUT_DIM + nn], v1);
        }
    }
}

// ---------------------------------------------------------------------------
// launch
// ---------------------------------------------------------------------------
extern "C" void kernel_launch(void* const* d_in, const int* in_sizes, int n_in,
                              void* d_out, int out_size, void* d_ws, size_t ws_size,
                              hipStream_t stream)
{
    const float* x   = (const float*)d_in[0];
    const float* We  = (const float*)d_in[1];
    const float* be  = (const float*)d_in[2];
    const float* Wg1 = (const float*)d_in[3];
    const float* bg1 = (const float*)d_in[4];
    const float* Wg2 = (const float*)d_in[5];
    const float* bg2 = (const float*)d_in[6];

    float* out      = (float*)d_out;                       // [N, OUT_DIM]
    float* gate_out = out + (size_t)N_TOKENS * OUT_DIM;    // [N, 8]
    int*   top2_out = (int*)(gate_out + (size_t)N_TOKENS * NEXP);  // [N, 2]

    // workspace layout (ints/floats, contiguous)
    int*   counts   = (int*)d_ws;                 // 8
    int*   cursors  = counts + 8;                 // 8
    int*   off      = cursors + 8;                // 16 (9 used)
    int*   ws_top2i = off + 16;                   // 2N
    float* ws_top2w = (float*)(ws_top2i + 2 * N_TOKENS);   // 2N
    int*   entryTok = (int*)(ws_top2w + 2 * N_TOKENS);     // ENTRY_CAP
    float* entryW   = (float*)(entryTok + ENTRY_CAP);      // ENTRY_CAP

    // zero counts/cursors/offsets, entry arrays, and the accumulated output
    hipMemsetAsync(counts, 0, 32 * sizeof(int), stream);
    hipMemsetAsync(entryTok, 0, (size_t)ENTRY_CAP * 2 * sizeof(float), stream);
    hipMemsetAsync(out, 0, (size_t)N_TOKENS * OUT_DIM * sizeof(float), stream);

    // 1) gating + top-2
    moe_gate_kernel<<<N_TOKENS, 64, 0, stream>>>(x, Wg1, bg1, Wg2, bg2,
                                                 gate_out, top2_out,
                                                 ws_top2i, ws_top2w, counts);
    // 2) segment offsets
    moe_offsets_kernel<<<1, 32, 0, stream>>>(counts, off);
    // 3) scatter into per-expert segments
    moe_scatter_kernel<<<(N_TOKENS + 255) / 256, 256, 0, stream>>>(
        ws_top2i, ws_top2w, off, cursors, entryTok, entryW);
    // 4) routed WMMA GEMM + combine
    dim3 grid(MAX_TILES, OUT_DIM / 256, 1);
    moe_expert_wmma_kernel<<<grid, 128, 0, stream>>>(x, We, be, off,
                                                     entryTok, entryW, out);
}